// CAGKE_learnable_minmax_54348516164262
// MI455X (gfx1250) — compile-verified
//
#include <hip/hip_runtime.h>

typedef float v2f __attribute__((ext_vector_type(2)));
typedef float v8f __attribute__((ext_vector_type(8)));

#define T_LEN 8192
#define D_LEN 128
#define RAD 56                 // Gaussian truncation radius (Kbar < 1e-40 beyond)
#define KSPAN 128              // k-window span = 2*RAD + 16, 32 WMMA K=4 steps
#define MPAD_LO 57             // zero padding in front of mask (covers index -57, keeps pairs 8B-aligned)
#define MPAD_HI 128            // zero padding behind mask (covers index 8246)
#define KREV_LEN 160           // reversed kernel table, indices 0..142 used
#define INV_SQRT_2PI 0.39894228f
#define NOISE_SIGMA 0.01f

__global__ __launch_bounds__(256)
void cagke_minmax_kernel(const float* __restrict__ X,
                         const float* __restrict__ weight,
                         const float* __restrict__ sigma_min,
                         const float* __restrict__ sigma_max,
                         const float* __restrict__ noise,
                         float* __restrict__ out)
{
    __shared__ __attribute__((aligned(16))) float s_maskp[MPAD_LO + T_LEN + MPAD_HI];
    __shared__ __attribute__((aligned(16))) float s_psedu[T_LEN];
    __shared__ __attribute__((aligned(16))) float s_kbr[KREV_LEN]; // r[idx] = Kbar(71 - idx)
    __shared__ float s_w[D_LEN];
    __shared__ float s_p[D_LEN];                                   // softmax(weight)
    __shared__ float s_red[64];
    __shared__ float s_rmin[256];
    __shared__ float s_rmax[256];

    const int tid = threadIdx.x;

    __builtin_prefetch(X + tid * 32, 0, 0);
    __builtin_prefetch(noise + tid * 32, 0, 0);

    // ---- Phase 1a: binarized, zero-padded mask into LDS; load weights ----
    for (int idx = tid; idx < MPAD_LO + T_LEN + MPAD_HI; idx += 256) {
        const int m = idx - MPAD_LO;
        float v = 0.0f;
        if (m >= 0 && m < T_LEN) v = (X[m] > 0.5f) ? 1.0f : 0.0f;
        s_maskp[idx] = v;
    }
    if (tid < D_LEN) s_w[tid] = weight[tid];
    __syncthreads();

    // ---- Phase 1b: softmax(weight) over D=128 ----
    if (tid < 64) s_red[tid] = fmaxf(s_w[tid], s_w[tid + 64]);
    __syncthreads();
    for (int s = 32; s > 0; s >>= 1) {
        if (tid < s) s_red[tid] = fmaxf(s_red[tid], s_red[tid + s]);
        __syncthreads();
    }
    const float wmax = s_red[0];
    __syncthreads();
    if (tid < D_LEN) s_p[tid] = expf(s_w[tid] - wmax);
    __syncthreads();
    if (tid < 64) s_red[tid] = s_p[tid] + s_p[tid + 64];
    __syncthreads();
    for (int s = 32; s > 0; s >>= 1) {
        if (tid < s) s_red[tid] += s_red[tid + s];
        __syncthreads();
    }
    const float wsum = s_red[0];
    __syncthreads();
    if (tid < D_LEN) s_p[tid] = s_p[tid] * (1.0f / wsum);
    __syncthreads();

    // ---- Phase 1c: REVERSED zero-padded combined kernel: r[idx] = Kbar(71 - idx) ----
    {
        const float smin = sigma_min[0];
        const float smax = sigma_max[0];
        if (tid < KREV_LEN) {
            float acc = 0.0f;
            const int delta = 71 - tid;              // valid only for |delta| <= 56
            if (delta >= -RAD && delta <= RAD) {
                const float d2 = (float)(delta * delta);
                for (int d = 0; d < D_LEN; ++d) {
                    float s = fabsf(smin + (float)d * (smax - smin) * (1.0f / (D_LEN - 1)));
                    acc += s_p[d] * (INV_SQRT_2PI / s) * expf(-d2 / (2.0f * s * s));
                }
            }
            s_kbr[tid] = acc;                        // out-of-range entries exactly zero
        }
    }
    __syncthreads();

    // ---- Phase 2: im2col GEMM conv via V_WMMA_F32_16X16X4_F32 ----
    // psedu[base + 16*i + n] = sum_ko mask[base + 16*i + ko - 57] * Kbar(n + 56 - ko)
    // A pair:  mrow[4ks], mrow[4ks+1]      (ascending, 8B-aligned)
    // B pair:  browr[4ks], browr[4ks+1]    (ascending via reversed table -> no swap movs)
    const int wave = tid >> 5;
    const int lane = tid & 31;
    const int half = lane >> 4;      // lanes 0-15: K=0,1  lanes 16-31: K=2,3
    const int l15  = lane & 15;

    for (int q = 0; q < 4; ++q) {
        const int base = (wave + 8 * q) * 256;
        const float* mrow  = s_maskp + base + 16 * l15 + 2 * half;  // (MPAD_LO-57)==0
        const float* browr = s_kbr + 15 - l15 + 2 * half;
        v8f c = {0.f, 0.f, 0.f, 0.f, 0.f, 0.f, 0.f, 0.f};
        #pragma unroll
        for (int ks = 0; ks < 32; ++ks) {
            v2f a, b;
            a.x = mrow[4 * ks];           // A[row=l15][K=ko0]
            a.y = mrow[4 * ks + 1];       // A[row=l15][K=ko0+1]
            b.x = browr[4 * ks];          // B[K=ko0][col=l15]   = Kbar(l15+56-ko0)
            b.y = browr[4 * ks + 1];      // B[K=ko0+1][col=l15] = Kbar(l15+56-ko0-1)
            c = __builtin_amdgcn_wmma_f32_16x16x4_f32(
                    false, a, false, b, (short)0, c, false, false);
        }
        // C/D layout: VGPR v -> M = v + 8*half, N = l15
        #pragma unroll
        for (int v = 0; v < 8; ++v) {
            const int i = v + 8 * half;
            const int t = base + 16 * i + l15;
            s_psedu[t] = c[v] + NOISE_SIGMA * noise[t];
        }
    }
    __syncthreads();

    // ---- Phase 3: global min/max + normalize ----
    float lo = 3.4028235e38f, hi = -3.4028235e38f;
    for (int idx = tid; idx < T_LEN; idx += 256) {
        const float v = s_psedu[idx];
        lo = fminf(lo, v);
        hi = fmaxf(hi, v);
    }
    s_rmin[tid] = lo;
    s_rmax[tid] = hi;
    __syncthreads();
    for (int s = 128; s > 0; s >>= 1) {
        if (tid < s) {
            s_rmin[tid] = fminf(s_rmin[tid], s_rmin[tid + s]);
            s_rmax[tid] = fmaxf(s_rmax[tid], s_rmax[tid + s]);
        }
        __syncthreads();
    }
    const float gmin = s_rmin[0];
    const float inv  = 1.0f / (s_rmax[0] - gmin);
    for (int idx = tid; idx < T_LEN; idx += 256)
        out[idx] = (s_psedu[idx] - gmin) * inv;
}

extern "C" void kernel_launch(void* const* d_in, const int* in_sizes, int n_in,
                              void* d_out, int out_size, void* d_ws, size_t ws_size,
                              hipStream_t stream) {
    const float* X     = (const float*)d_in[0];
    const float* w     = (const float*)d_in[1];
    const float* smin  = (const float*)d_in[2];
    const float* smax  = (const float*)d_in[3];
    const float* nz    = (const float*)d_in[4];
    float* out         = (float*)d_out;
    (void)in_sizes; (void)n_in; (void)out_size; (void)d_ws; (void)ws_size;
    cagke_minmax_kernel<<<1, 256, 0, stream>>>(X, w, smin, smax, nz, out);
}